// Transfer0_1_73332271612005
// MI455X (gfx1250) — compile-verified
//
#include <hip/hip_runtime.h>
#include <hip/hip_bf16.h>

// ---------------------------------------------------------------------------
// Problem constants (fixed by the reference)
// ---------------------------------------------------------------------------
#define Nn  262144   // source rows
#define Hh  128      // hidden units
#define Dd  65536    // domains
#define ENn 1048576  // node_map edges
#define EDd 524288   // domain_map edges (messages)

#define LDSW 132     // padded LDS row stride (floats): 132 % 64 == 4 -> conflict-free b64 frag loads
#define OTW  260     // padded LDS out stride for the 256-wide message tile (260 % 64 == 4)

typedef __attribute__((ext_vector_type(2))) float v2f;
typedef __attribute__((ext_vector_type(8))) float v8f;

// ---------------------------------------------------------------------------
// One 16x16 fp32 WMMA output tile:  D = Xl(16x128) * W.T(strip of 16 rows)
// Xl: LDS, 16 rows x LDSW floats.  Wrow: global, points at W + n0*128 (row-major [*,128]).
// ISA layouts (7.12.2): A 16x4: M=lane&15, VGPR0/1 = K pair {2h, 2h+1};  B 4x16 symmetric
// with N=lane&15;  D: VGPR r -> row 8h+r, col lane&15.
// ---------------------------------------------------------------------------
__device__ __forceinline__ v8f wmma_rowtile_f32(const float* __restrict__ Xl,
                                                const float* __restrict__ Wrow,
                                                int m, int h) {
  v8f acc = {0.f, 0.f, 0.f, 0.f, 0.f, 0.f, 0.f, 0.f};
#pragma unroll
  for (int k0 = 0; k0 < Hh; k0 += 4) {
    const int k = k0 + 2 * h;
    const v2f a = *(const v2f*)(Xl + m * LDSW + k);            // A frag (LDS, bank-clean)
    const v2f b = *(const v2f*)(Wrow + (long)m * Hh + k);      // B frag = W[n0+m][k..k+1]
    acc = __builtin_amdgcn_wmma_f32_16x16x4_f32(false, a, false, b,
                                                (short)0, acc, false, false);
  }
  return acc;
}

// ---------------------------------------------------------------------------
// Kernel A: segdom[d] += x[r] for sorted domain_indicator (run-length + atomics at run ends).
// One wave per 16 consecutive rows; lane owns 4 columns.
// ---------------------------------------------------------------------------
__global__ __launch_bounds__(256) void seg_rows_kernel(const float* __restrict__ x,
                                                       const int* __restrict__ dom,
                                                       float* __restrict__ segdom) {
  const int  lane = threadIdx.x & 31;
  const long wave = ((long)blockIdx.x * blockDim.x + threadIdx.x) >> 5;
  const long r0   = wave * 16;
  float a0 = 0.f, a1 = 0.f, a2 = 0.f, a3 = 0.f;
  int cur = dom[r0];
#pragma unroll 4
  for (int i = 0; i < 16; ++i) {
    const long r = r0 + i;
    const int  d = dom[r];
    if (d != cur) {
      float* p = segdom + (long)cur * Hh + lane * 4;
      atomicAdd(p + 0, a0); atomicAdd(p + 1, a1);
      atomicAdd(p + 2, a2); atomicAdd(p + 3, a3);
      a0 = a1 = a2 = a3 = 0.f;
      cur = d;
    }
    const float4 v = *(const float4*)(x + r * Hh + lane * 4);
    a0 += v.x; a1 += v.y; a2 += v.z; a3 += v.w;
  }
  float* p = segdom + (long)cur * Hh + lane * 4;
  atomicAdd(p + 0, a0); atomicAdd(p + 1, a1);
  atomicAdd(p + 2, a2); atomicAdd(p + 3, a3);
}

// ---------------------------------------------------------------------------
// Kernel B: segint[s] += x[node_map[e]] for sorted intersect_indicator (gather + run-length).
// x (134 MB) is L2-resident, so the gathers hit L2.
// ---------------------------------------------------------------------------
__global__ __launch_bounds__(256) void seg_edges_kernel(const float* __restrict__ x,
                                                        const int* __restrict__ nmap,
                                                        const int* __restrict__ isec,
                                                        float* __restrict__ segint) {
  const int  lane = threadIdx.x & 31;
  const long wave = ((long)blockIdx.x * blockDim.x + threadIdx.x) >> 5;
  const long e0   = wave * 16;
  float a0 = 0.f, a1 = 0.f, a2 = 0.f, a3 = 0.f;
  int cur = isec[e0];
#pragma unroll 4
  for (int i = 0; i < 16; ++i) {
    const long e = e0 + i;
    const int  s = isec[e];
    if (s != cur) {
      float* p = segint + (long)cur * Hh + lane * 4;
      atomicAdd(p + 0, a0); atomicAdd(p + 1, a1);
      atomicAdd(p + 2, a2); atomicAdd(p + 3, a3);
      a0 = a1 = a2 = a3 = 0.f;
      cur = s;
    }
    const long row = nmap[e];
    const float4 v = *(const float4*)(x + row * Hh + lane * 4);
    a0 += v.x; a1 += v.y; a2 += v.z; a3 += v.w;
  }
  float* p = segint + (long)cur * Hh + lane * 4;
  atomicAdd(p + 0, a0); atomicAdd(p + 1, a1);
  atomicAdd(p + 2, a2); atomicAdd(p + 3, a3);
}

// ---------------------------------------------------------------------------
// Kernel X: in-place row transform  buf[i,:] = buf[i,:] @ W.T   (W row-major [128,128]).
// 16-row tile per block; 8 waves, one 16-col strip each. Safe in-place (row i -> row i).
// ---------------------------------------------------------------------------
__global__ __launch_bounds__(256) void row_xform_kernel(float* __restrict__ buf,
                                                        const float* __restrict__ W) {
  __shared__ float Xl[16 * LDSW];
  __shared__ float Ot[16 * LDSW];
  const int  t    = threadIdx.x;
  const int  lane = t & 31;
  const int  w    = t >> 5;
  const long r0   = (long)blockIdx.x * 16;

#pragma unroll
  for (int j = 0; j < 2; ++j) {                      // 512 float4: stage 16x128 rows
    const int f = t + 256 * j, row = f >> 5, c4 = f & 31;
    *(float4*)(Xl + row * LDSW + c4 * 4) =
        *(const float4*)(buf + (r0 + row) * Hh + c4 * 4);
  }
  __syncthreads();

  const int m = lane & 15, h = lane >> 4, n0 = w * 16;
  const v8f acc = wmma_rowtile_f32(Xl, W + (long)n0 * Hh, m, h);
#pragma unroll
  for (int r = 0; r < 8; ++r)                        // D layout: row 8h+r, col m
    Ot[(8 * h + r) * LDSW + n0 + m] = acc[r];
  __syncthreads();

#pragma unroll
  for (int j = 0; j < 2; ++j) {                      // coalesced write-back (in place)
    const int f = t + 256 * j, row = f >> 5, c4 = f & 31;
    *(float4*)(buf + (r0 + row) * Hh + c4 * 4) =
        *(const float4*)(Ot + row * LDSW + c4 * 4);
  }
}

// ---------------------------------------------------------------------------
// Kernel C: fused message build.  Per block: 16 messages ->
//   T = x[dom_src] @ W_x.T (16x256 via 2 WMMA strips/wave)
//   yc = y_sum[dom_dst] + y_int           (16x128, added on store)
//   out rows [2*e0 .. 2*e0+31] = msg (raw), plus per-column sum/sumsq atomics.
// ---------------------------------------------------------------------------
__global__ __launch_bounds__(256) void msg_kernel(const float* __restrict__ x,
                                                  const float* __restrict__ Wx,     // [256,128]
                                                  const float* __restrict__ ysum,   // segdom (xformed)
                                                  const float* __restrict__ yint,   // segint (xformed)
                                                  const int* __restrict__ dsrc,
                                                  const int* __restrict__ ddst,
                                                  float* __restrict__ out,          // [2*ED,128]
                                                  float* __restrict__ statS,
                                                  float* __restrict__ statQ) {
  __shared__ float Ax[16 * LDSW];
  __shared__ float Yc[16 * LDSW];
  __shared__ float Ot[16 * OTW];
  __shared__ float RedS[128];
  __shared__ float RedQ[128];

  const int  t    = threadIdx.x;
  const int  lane = t & 31;
  const int  w    = t >> 5;
  const long e0   = (long)blockIdx.x * 16;

#pragma unroll
  for (int j = 0; j < 2; ++j) {                      // gather x[dom_src] rows
    const int f = t + 256 * j, row = f >> 5, c4 = f & 31;
    const long src = dsrc[e0 + row];
    *(float4*)(Ax + row * LDSW + c4 * 4) =
        *(const float4*)(x + src * Hh + c4 * 4);
  }
#pragma unroll
  for (int j = 0; j < 2; ++j) {                      // yc = y_sum[dom_dst] + y_int
    const int f = t + 256 * j, row = f >> 5, c4 = f & 31;
    const long dd = ddst[e0 + row];
    const float4 a = *(const float4*)(ysum + dd * Hh + c4 * 4);
    const float4 b = *(const float4*)(yint + (e0 + row) * Hh + c4 * 4);
    float4 r4; r4.x = a.x + b.x; r4.y = a.y + b.y; r4.z = a.z + b.z; r4.w = a.w + b.w;
    *(float4*)(Yc + row * LDSW + c4 * 4) = r4;
  }
  __syncthreads();

  const int m = lane & 15, h = lane >> 4, n0 = w * 16;
  const v8f acc0 = wmma_rowtile_f32(Ax, Wx + (long)n0 * Hh, m, h);          // cols [n0,n0+16)
  const v8f acc1 = wmma_rowtile_f32(Ax, Wx + (long)(128 + n0) * Hh, m, h);  // cols [128+n0,...)
#pragma unroll
  for (int r = 0; r < 8; ++r) {
    const int row = 8 * h + r;
    const float y = Yc[row * LDSW + n0 + m];
    Ot[row * OTW + n0 + m]       = acc0[r] + y;      // msg half 0
    Ot[row * OTW + 128 + n0 + m] = acc1[r] + y;      // msg half 1
  }
  __syncthreads();

  // ---- per-column stats (combine halves in LDS, then one atomic pair per column) ----
  {
    const int c = t & 127, half = t >> 7;
    float s = 0.f, q = 0.f;
#pragma unroll 4
    for (int i = 0; i < 16; ++i) {
      const float v = Ot[i * OTW + half * 128 + c];
      s += v; q += v * v;
    }
    if (half == 1) { RedS[c] = s; RedQ[c] = q; }
    __syncthreads();
    if (half == 0) {
      atomicAdd(statS + c, s + RedS[c]);
      atomicAdd(statQ + c, q + RedQ[c]);
    }
  }

  // ---- contiguous 32x128 store: global row 2*e0+rr  <->  Ot[rr>>1][(rr&1)*128 + c] ----
#pragma unroll
  for (int j = 0; j < 4; ++j) {
    const int f = t + 256 * j;                       // 1024 float4
    const int rr = f >> 5, c4 = f & 31;
    const int i = rr >> 1, hf = rr & 1;
    *(float4*)(out + (2 * e0 + rr) * Hh + c4 * 4) =
        *(const float4*)(Ot + i * OTW + hf * 128 + c4 * 4);
  }
}

// ---------------------------------------------------------------------------
// Kernel D: BatchNorm affine folding:  a = gamma*rsqrt(var+eps), b = beta - mean*a
// ---------------------------------------------------------------------------
__global__ void finalize_stats_kernel(const float* __restrict__ s, const float* __restrict__ q,
                                      const float* __restrict__ g, const float* __restrict__ b,
                                      float* __restrict__ ab) {
  const int   c    = threadIdx.x;                    // 128 threads
  const float inv  = 1.0f / (float)(2 * EDd);
  const float mean = s[c] * inv;
  const float var  = fmaxf(q[c] * inv - mean * mean, 0.f);
  const float aa   = g[c] * rsqrtf(var + 1e-5f);
  ab[c]       = aa;
  ab[128 + c] = b[c] - mean * aa;
}

// ---------------------------------------------------------------------------
// Kernel E: in-place  out = relu(out*a[c] + b[c])  — streaming float4 pass.
// ---------------------------------------------------------------------------
__global__ __launch_bounds__(256) void norm_relu_kernel(float* __restrict__ out,
                                                        const float* __restrict__ ab) {
#pragma unroll
  for (int j = 0; j < 4; ++j) {
    const long i  = (long)blockIdx.x * 1024 + j * 256 + threadIdx.x;  // float4 index
    const int  c0 = ((int)(i & 31)) * 4;             // 32 float4 per 128-col row
    float4 v = *(float4*)(out + i * 4);
    const float4 a = *(const float4*)(ab + c0);
    const float4 b = *(const float4*)(ab + 128 + c0);
    v.x = fmaxf(fmaf(v.x, a.x, b.x), 0.f);
    v.y = fmaxf(fmaf(v.y, a.y, b.y), 0.f);
    v.z = fmaxf(fmaf(v.z, a.z, b.z), 0.f);
    v.w = fmaxf(fmaf(v.w, a.w, b.w), 0.f);
    *(float4*)(out + i * 4) = v;
  }
}

// ---------------------------------------------------------------------------
// Launch
// ---------------------------------------------------------------------------
extern "C" void kernel_launch(void* const* d_in, const int* in_sizes, int n_in,
                              void* d_out, int out_size, void* d_ws, size_t ws_size,
                              hipStream_t stream) {
  (void)in_sizes; (void)n_in; (void)out_size; (void)ws_size;

  const float* x      = (const float*)d_in[0];
  const float* W_sum  = (const float*)d_in[1];
  const float* W_int  = (const float*)d_in[2];
  const float* W_x    = (const float*)d_in[3];
  const float* bn_w   = (const float*)d_in[4];
  const float* bn_b   = (const float*)d_in[5];
  const int*   dom    = (const int*)d_in[6];
  const int*   nmap   = (const int*)d_in[7];
  const int*   isec   = (const int*)d_in[8];
  const int*   dsrc   = (const int*)d_in[9];
  const int*   ddst   = (const int*)d_in[10];
  float*       out    = (float*)d_out;

  // Workspace layout (~302 MB):  segdom [D,H] | segint [ED,H] | stats (512 f32)
  char*  ws      = (char*)d_ws;
  float* segdom  = (float*)ws;
  float* segint  = (float*)(ws + (size_t)Dd * Hh * sizeof(float));
  float* stats   = (float*)(ws + (size_t)Dd * Hh * sizeof(float)
                               + (size_t)EDd * Hh * sizeof(float));
  float* statS   = stats;         // [128]
  float* statQ   = stats + 128;   // [128]
  float* ab      = stats + 256;   // [256]: a | b

  hipMemsetAsync(segdom, 0, (size_t)Dd * Hh * sizeof(float), stream);
  hipMemsetAsync(segint, 0, (size_t)EDd * Hh * sizeof(float), stream);
  hipMemsetAsync(statS, 0, 256 * sizeof(float), stream);

  // 1) segment sums (sorted indicators, run-length + atomics)
  seg_rows_kernel <<<Nn  / 16 / 8, 256, 0, stream>>>(x, dom, segdom);
  seg_edges_kernel<<<ENn / 16 / 8, 256, 0, stream>>>(x, nmap, isec, segint);

  // 2) in-place row GEMMs: segdom <- segdom@W_sum.T ; segint <- segint@W_int.T
  row_xform_kernel<<<Dd  / 16, 256, 0, stream>>>(segdom, W_sum);
  row_xform_kernel<<<EDd / 16, 256, 0, stream>>>(segint, W_int);

  // 3) fused message build (raw msg -> d_out) + column stats
  msg_kernel<<<EDd / 16, 256, 0, stream>>>(x, W_x, segdom, segint, dsrc, ddst,
                                           out, statS, statQ);

  // 4) BatchNorm fold + 5) normalize+ReLU in place
  finalize_stats_kernel<<<1, 128, 0, stream>>>(statS, statQ, bn_w, bn_b, ab);
  norm_relu_kernel<<<(int)(((long)2 * EDd * Hh / 4) / 1024), 256, 0, stream>>>(out, ab);
}